// ViterbiBatchSize1_24361054503464
// MI455X (gfx1250) — compile-verified
//
#include <hip/hip_runtime.h>

// ---------------- problem constants ----------------
#define SEQ_LEN   32768
#define NUM_TAGS  512
#define START_IDX 510
#define END_IDX   511
#define NEG_FILL  (-10000.0f)

// ---------------- decomposition ----------------
#define NUM_WGS      32                      // workgroups cooperating via grid barrier
#define ROWS_PER_WG  (NUM_TAGS / NUM_WGS)    // 16 output rows per WG
#define THREADS      512                     // 16 wave32s per WG -> 1 row per wave

typedef unsigned int   uint32;
typedef unsigned short u16;
typedef uint32 v4u __attribute__((ext_vector_type(4)));
typedef int    v8i __attribute__((ext_vector_type(8)));
typedef int    v4i __attribute__((ext_vector_type(4)));

// d_ws layout
#define WS_COUNTER_OFF 0        // 1 uint (grid-barrier monotonic counter)
#define WS_ALPHA_OFF   256      // 2 * 512 floats (double-buffered alphas)
#define WS_BP_OFF      8192     // SEQ_LEN * NUM_TAGS u16 backpointers (32 MB)

// Reset the grid-barrier counter every launch (graph replay safe).
__global__ void viterbi_init(uint32* __restrict__ counter) {
    if (threadIdx.x == 0) *counter = 0u;
}

// Forward pass: max-plus matvec per step, rows partitioned across 32 WGs,
// grid-synchronized once per step via a monotonic atomic counter.
__global__ __launch_bounds__(THREADS) void viterbi_forward(
    const float* __restrict__ unary,      // [SEQ_LEN, NUM_TAGS]
    const float* __restrict__ trans,      // [NUM_TAGS, NUM_TAGS]
    float*       __restrict__ alphaBuf,   // [2, NUM_TAGS]
    u16*         __restrict__ bp,         // [SEQ_LEN, NUM_TAGS]
    uint32*      __restrict__ counter)
{
    __shared__ float lds_tr[ROWS_PER_WG * NUM_TAGS];  // 32 KB: this WG's trans slice
    __shared__ float lds_alpha[NUM_TAGS];             // 2 KB : current alphas

    const int tid     = threadIdx.x;
    const int wg      = blockIdx.x;
    const int wave    = tid >> 5;
    const int lane    = tid & 31;
    const int rowBase = wg * ROWS_PER_WG;

    // ---- stage trans[rowBase : rowBase+16, :] into LDS via Tensor Data Mover ----
#if defined(__HIP_DEVICE_COMPILE__) && __has_builtin(__builtin_amdgcn_tensor_load_to_lds)
    if (wave == 0) {
        // Tensor DMA descriptor (D#): 2D tile 512x16 f32, contiguous into LDS.
        unsigned long long ga = (unsigned long long)(size_t)(trans + (size_t)rowBase * NUM_TAGS);
        uint32 ldsOff = (uint32)(size_t)(&lds_tr[0]);     // flat LDS addr low 32 bits
        v4u g0 = { 1u,                                    // count=1 (valid user D#)
                   ldsOff,                                // lds_addr (bytes)
                   (uint32)(ga & 0xFFFFFFFFull),          // global_addr[31:0]
                   ((uint32)((ga >> 32) & 0x01FFFFFFull)) // global_addr[56:32]
                     | (2u << 30) };                      // type=2 ("image")
        v8i g1 = { (int)(2u << 16),                       // data_size=2 -> 4 bytes
                   (int)((NUM_TAGS    & 0xFFFF) << 16),   // tensor_dim0 = 512
                   (int)((ROWS_PER_WG & 0xFFFF) << 16),   // tensor_dim1 = 16
                   (int)((NUM_TAGS    & 0xFFFF) << 16),   // tile_dim0   = 512
                   (int)(ROWS_PER_WG),                    // tile_dim1   = 16
                   (int)(NUM_TAGS),                       // tensor_dim0_stride = 512
                   0, 0 };
        v4i gz4 = { 0, 0, 0, 0 };                         // dims 2..4 unused (2D tile)
        v8i gz8 = { 0, 0, 0, 0, 0, 0, 0, 0 };             // extension group (unused)
        __builtin_amdgcn_tensor_load_to_lds(g0, g1, gz4, gz4, gz8, 0);
        __builtin_amdgcn_s_wait_tensorcnt(0);
    }
#else
    for (int idx = tid; idx < ROWS_PER_WG * NUM_TAGS; idx += THREADS)
        lds_tr[idx] = trans[(size_t)rowBase * NUM_TAGS + idx];
#endif

    // alphas0: 0 at START_IDX, NEG_FILL elsewhere
    lds_alpha[tid] = (tid == START_IDX) ? 0.0f : NEG_FILL;
    __syncthreads();

    for (int t = 0; t < SEQ_LEN; ++t) {
        // prefetch unary rows ~16 steps ahead (global_prefetch_b8)
        if (wave == 0 && lane < ROWS_PER_WG && (t + 16) < SEQ_LEN)
            __builtin_prefetch(&unary[(size_t)(t + 16) * NUM_TAGS + rowBase + lane], 0, 1);

        // one wave per output row: row i = rowBase + wave.
        // Vectorized LDS reads: lane handles j in {4*lane+128k .. +3}, k=0..3
        // (ds_load_b128, conflict-free; per-lane j strictly ascending).
        const int row = wave;
        const float4* __restrict__ trRow4 = (const float4*)&lds_tr[row * NUM_TAGS];
        const float4* __restrict__ al4    = (const float4*)lds_alpha;
        float m  = -3.0e38f;
        int   mi = 0;
        #pragma unroll
        for (int k = 0; k < NUM_TAGS / 128; ++k) {
            const int    q  = (k << 5) + lane;            // float4 slot
            const float4 a4 = al4[q];
            const float4 t4 = trRow4[q];
            const int    jb = q << 2;                     // base tag index
            float s;
            s = a4.x + t4.x; if (s > m) { m = s; mi = jb;     }
            s = a4.y + t4.y; if (s > m) { m = s; mi = jb + 1; }
            s = a4.z + t4.z; if (s > m) { m = s; mi = jb + 2; }
            s = a4.w + t4.w; if (s > m) { m = s; mi = jb + 3; }
        }
        // wave32 (value, index) max-reduction; ties -> smallest j (jnp.argmax semantics)
        #pragma unroll
        for (int off = 16; off >= 1; off >>= 1) {
            const float om = __shfl_xor(m,  off, 32);
            const int   oi = __shfl_xor(mi, off, 32);
            if (om > m || (om == m && oi < mi)) { m = om; mi = oi; }
        }
        if (lane == 0) {
            const int grow = rowBase + row;
            alphaBuf[((t + 1) & 1) * NUM_TAGS + grow] =
                m + unary[(size_t)t * NUM_TAGS + grow];
            bp[(size_t)t * NUM_TAGS + grow] = (u16)mi;
        }

        // ---- grid barrier: make this step's alphas visible, then sync ----
        __threadfence();                 // release this thread's global stores
        __syncthreads();
        if (tid == 0) {
            atomicAdd(counter, 1u);
            const uint32 target = (uint32)(t + 1) * (uint32)NUM_WGS;
            while (__hip_atomic_load(counter, __ATOMIC_ACQUIRE,
                                     __HIP_MEMORY_SCOPE_AGENT) < target)
                __builtin_amdgcn_s_sleep(2);
        }
        __syncthreads();

        // reload full alpha vector (device-scope load: skip stale WGP$)
        lds_alpha[tid] = __hip_atomic_load(
            &alphaBuf[((t + 1) & 1) * NUM_TAGS + tid],
            __ATOMIC_RELAXED, __HIP_MEMORY_SCOPE_AGENT);
        __syncthreads();
    }
}

// Terminal scores + sequential backtrace. Final alphas live in buffer 0
// (last step t=32767 wrote buffer (t+1)&1 == 0).
__global__ __launch_bounds__(NUM_TAGS) void viterbi_traceback(
    const float* __restrict__ trans,
    const float* __restrict__ alphaBuf,
    const u16*   __restrict__ bp,
    float*       __restrict__ out)        // [SEQ_LEN] path + [1] score
{
    __shared__ float vals[NUM_TAGS];
    const int tid = threadIdx.x;
    vals[tid] = alphaBuf[tid] + trans[(size_t)END_IDX * NUM_TAGS + tid];
    __syncthreads();
    if (tid == 0) {
        float m  = vals[0];
        int   mi = 0;
        for (int j = 1; j < NUM_TAGS; ++j)
            if (vals[j] > m) { m = vals[j]; mi = j; }     // first-argmax
        out[SEQ_LEN] = m;                                  // path_score
        int tag = mi;
        for (int t = SEQ_LEN - 1; t >= 0; --t) {
            out[t] = (float)tag;                           // path[t]
            tag = (int)bp[(size_t)t * NUM_TAGS + tag];
        }
    }
}

extern "C" void kernel_launch(void* const* d_in, const int* in_sizes, int n_in,
                              void* d_out, int out_size, void* d_ws, size_t ws_size,
                              hipStream_t stream) {
    const float* unary = (const float*)d_in[0];   // [32768, 1, 512] f32
    const float* trans = (const float*)d_in[1];   // [1, 512, 512]   f32
    (void)d_in; (void)in_sizes; (void)n_in; (void)out_size; (void)ws_size;

    char*   ws       = (char*)d_ws;
    uint32* counter  = (uint32*)(ws + WS_COUNTER_OFF);
    float*  alphaBuf = (float*) (ws + WS_ALPHA_OFF);
    u16*    bp       = (u16*)   (ws + WS_BP_OFF);
    float*  out      = (float*)d_out;

    hipLaunchKernelGGL(viterbi_init, dim3(1), dim3(64), 0, stream, counter);
    hipLaunchKernelGGL(viterbi_forward, dim3(NUM_WGS), dim3(THREADS), 0, stream,
                       unary, trans, alphaBuf, bp, counter);
    hipLaunchKernelGGL(viterbi_traceback, dim3(1), dim3(NUM_TAGS), 0, stream,
                       trans, alphaBuf, bp, out);
}